// RPNPostProcessor_39204461478681
// MI455X (gfx1250) — compile-verified
//
#include <hip/hip_runtime.h>
#include <stdint.h>

// ---------------- problem constants (match reference) ----------------
#define N_IMG   8
#define A_      3
#define H_      272
#define W_      400
#define HW_     (H_ * W_)          // 108800
#define M_      (A_ * HW_)         // 326400 candidates per image
#define PRE_NMS 2000
#define POST_NMS 1000
#define SORT_N  2048               // pow2 pad for bitonic sort
#define NMS_THRESH 0.7f
#define NEG_F   (-1e30f)
#define IM_W_F  1600.0f
#define IM_H_F  1088.0f
#define BBOX_CLIP 4.135166556742356f   // log(1000/16)

// histogram select
#define BIN_BITS  14
#define NBINS     (1 << BIN_BITS)      // 16384
#define BIN_SHIFT (32 - BIN_BITS)      // 18

// ---------------- monotone float <-> uint32 key ----------------
__device__ __forceinline__ uint32_t fkey(float f) {
    uint32_t u = __float_as_uint(f);
    return (u & 0x80000000u) ? ~u : (u | 0x80000000u);
}
__device__ __forceinline__ float keyf(uint32_t k) {
    uint32_t u = (k & 0x80000000u) ? (k & 0x7fffffffu) : ~k;
    return __uint_as_float(u);
}

// strict total order: key descending, index ascending on ties
__device__ __forceinline__ bool pair_before(uint2 a, uint2 b) {
    return (a.x > b.x) || (a.x == b.x && a.y < b.y);
}

// ---------------- CDNA5 async global->LDS staging ----------------
__device__ __forceinline__ void async_load_lds_b64(uint32_t lds_off, uint64_t gaddr) {
    // GLOBAL_LOAD_ASYNC_TO_LDS_B64 (opcode 97): each lane copies 8 bytes from global memory to its LDS address
    asm volatile("global_load_async_to_lds_b64 %0, %1, off"
                 :: "v"(lds_off), "v"(gaddr) : "memory");
}
__device__ __forceinline__ void wait_asynccnt0() {
    asm volatile("s_wait_asynccnt 0" ::: "memory");
}

// ---------------- k0: zero scratch counters/histograms ----------------
__global__ void rpn_k0_zero(uint32_t* __restrict__ p, int n) {
    for (int i = blockIdx.x * blockDim.x + threadIdx.x; i < n; i += gridDim.x * blockDim.x)
        p[i] = 0u;
}

// ---------------- k1: per-image 16K-bin histogram of score keys ----------------
__global__ void rpn_k1_hist(const float* __restrict__ obj, uint32_t* __restrict__ hist) {
    __shared__ uint32_t lh[NBINS];                       // 64 KB LDS
    const int n = blockIdx.y;
    for (int b = threadIdx.x; b < NBINS; b += blockDim.x) lh[b] = 0u;
    __syncthreads();
    const float* o = obj + (size_t)n * M_;
    for (int j = blockIdx.x * blockDim.x + threadIdx.x; j < M_; j += gridDim.x * blockDim.x)
        atomicAdd(&lh[fkey(o[j]) >> BIN_SHIFT], 1u);
    __syncthreads();
    uint32_t* gh = hist + (size_t)n * NBINS;
    for (int b = threadIdx.x; b < NBINS; b += blockDim.x) {
        uint32_t c = lh[b];
        if (c) atomicAdd(&gh[b], c);
    }
}

// ---------------- k2: find threshold bin for top-PRE_NMS ----------------
__global__ void rpn_k2_thresh(const uint32_t* __restrict__ hist, uint32_t* __restrict__ thr) {
    __shared__ uint32_t csum[256];
    const int n = blockIdx.x;
    const uint32_t* gh = hist + (size_t)n * NBINS;
    const int CH = NBINS / 256;                          // 64 bins per thread
    uint32_t s = 0;
    for (int b = threadIdx.x * CH; b < (threadIdx.x + 1) * CH; ++b) s += gh[b];
    csum[threadIdx.x] = s;
    __syncthreads();
    if (threadIdx.x == 0) {
        uint32_t above = 0;
        int c;
        for (c = 255; c >= 0; --c) {
            if (above + csum[c] >= PRE_NMS) break;
            above += csum[c];
        }
        if (c < 0) c = 0;
        int thrBin = c * CH;
        for (int b = (c + 1) * CH - 1; b >= c * CH; --b) {
            if (above + gh[b] >= PRE_NMS) { thrBin = b; break; }
            above += gh[b];
        }
        thr[n * 4 + 0] = (uint32_t)thrBin;               // threshold bin
        thr[n * 4 + 1] = PRE_NMS - above;                // needFromBin
        thr[n * 4 + 2] = above;                          // countAbove (bins > thrBin)
        thr[n * 4 + 3] = 0u;
    }
}

// ---------------- k3: compact top-PRE_NMS (key, candidate index m) ----------------
__global__ void rpn_k3_select(const float* __restrict__ obj, const uint32_t* __restrict__ thr,
                              uint32_t* __restrict__ cnt, uint2* __restrict__ pairs) {
    const int n = blockIdx.y;
    const uint32_t thrBin = thr[n * 4 + 0];
    const uint32_t need   = thr[n * 4 + 1];
    const uint32_t above  = thr[n * 4 + 2];
    const float* o = obj + (size_t)n * M_;
    uint2* pp = pairs + (size_t)n * SORT_N;
    for (int j = blockIdx.x * blockDim.x + threadIdx.x; j < M_; j += gridDim.x * blockDim.x) {
        uint32_t key = fkey(o[j]);
        uint32_t bin = key >> BIN_SHIFT;
        if (bin < thrBin) continue;
        // j = a*HW + (h*W + w)  ->  m = (h*W + w)*A + a  (reference transpose order)
        uint32_t m = (uint32_t)(j % HW_) * A_ + (uint32_t)(j / HW_);
        if (bin > thrBin) {
            uint32_t pos = atomicAdd(&cnt[n * 2 + 0], 1u);
            if (pos < SORT_N) pp[pos] = make_uint2(key, m);
        } else {
            uint32_t t = atomicAdd(&cnt[n * 2 + 1], 1u);
            if (t < need) {
                uint32_t pos = above + t;
                if (pos < SORT_N) pp[pos] = make_uint2(key, m);
            }
        }
    }
}

// ---------------- k4: async-stage -> bitonic sort -> decode -> greedy NMS ----------------
__global__ __launch_bounds__(1024)
void rpn_k4_sort_decode_nms(const float* __restrict__ breg, const float* __restrict__ anchors,
                            const uint2* __restrict__ pairs, float* __restrict__ out) {
    __shared__ uint2 spair[SORT_N];                      // 16 KB
    __shared__ float sx1[SORT_N], sy1[SORT_N], sx2[SORT_N], sy2[SORT_N]; // 32 KB
    __shared__ float ssc[SORT_N];                        // 8 KB
    __shared__ unsigned char srm[SORT_N];                // 2 KB
    __shared__ int kept;

    const int n   = blockIdx.x;
    const int tid = threadIdx.x;
    const uint2* pp = pairs + (size_t)n * SORT_N;

    // --- stage candidate pairs into LDS with CDNA5 async loads (ASYNCcnt) ---
    for (int p = tid; p < PRE_NMS; p += 1024) {
        uint32_t lds = (uint32_t)(uintptr_t)(const void*)&spair[p];
        uint64_t g   = (uint64_t)(uintptr_t)(const void*)&pp[p];
        async_load_lds_b64(lds, g);
    }
    for (int p = PRE_NMS + tid; p < SORT_N; p += 1024)
        spair[p] = make_uint2(0u, 0u);                   // sentinel sorts last
    wait_asynccnt0();
    __syncthreads();

    // --- bitonic sort, descending by key (idx-ascending tiebreak) ---
    for (int k = 2; k <= SORT_N; k <<= 1) {
        for (int j = k >> 1; j > 0; j >>= 1) {
            for (int i = tid; i < SORT_N; i += 1024) {
                int ixj = i ^ j;
                if (ixj > i) {
                    uint2 a = spair[i], b = spair[ixj];
                    bool sw = ((i & k) == 0) ? pair_before(b, a) : pair_before(a, b);
                    if (sw) { spair[i] = b; spair[ixj] = a; }
                }
            }
            __syncthreads();
        }
    }

    // --- decode + clip + min-size mask ---
    const float* anc   = anchors + (size_t)n * M_ * 4;
    const size_t bbase = (size_t)n * (A_ * 4) * HW_;
    for (int s = tid; s < SORT_N; s += 1024) {
        float X1 = 0.f, Y1 = 0.f, X2 = 0.f, Y2 = 0.f, sc = NEG_F;
        if (s < PRE_NMS) {
            uint2 e = spair[s];
            if (e.x != 0u) {
                float logit = keyf(e.x);
                float score = 1.f / (1.f + __expf(-logit));
                uint32_t m  = e.y;
                uint32_t a  = m % A_;
                uint32_t hw = m / A_;
                const float* ab = anc + (size_t)m * 4;
                float ax1 = ab[0], ay1 = ab[1], ax2 = ab[2], ay2 = ab[3];
                size_t c0 = bbase + (size_t)(a * 4) * HW_ + hw;
                float dx = breg[c0];
                float dy = breg[c0 + HW_];
                float dw = fminf(breg[c0 + 2 * (size_t)HW_], BBOX_CLIP);
                float dh = fminf(breg[c0 + 3 * (size_t)HW_], BBOX_CLIP);
                float wa = ax2 - ax1 + 1.f, ha = ay2 - ay1 + 1.f;
                float cx = ax1 + 0.5f * wa, cy = ay1 + 0.5f * ha;
                float pcx = dx * wa + cx,  pcy = dy * ha + cy;
                float pw  = __expf(dw) * wa, ph = __expf(dh) * ha;
                X1 = pcx - 0.5f * pw;        Y1 = pcy - 0.5f * ph;
                X2 = pcx + 0.5f * pw - 1.f;  Y2 = pcy + 0.5f * ph - 1.f;
                X1 = fminf(fmaxf(X1, 0.f), IM_W_F - 1.f);
                X2 = fminf(fmaxf(X2, 0.f), IM_W_F - 1.f);
                Y1 = fminf(fmaxf(Y1, 0.f), IM_H_F - 1.f);
                Y2 = fminf(fmaxf(Y2, 0.f), IM_H_F - 1.f);
                float bw = X2 - X1 + 1.f, bh = Y2 - Y1 + 1.f;
                sc = (bw >= 0.f && bh >= 0.f) ? score : NEG_F;
            }
        }
        sx1[s] = X1; sy1[s] = Y1; sx2[s] = X2; sy2[s] = Y2; ssc[s] = sc;
        srm[s] = 0;
    }
    if (tid == 0) kept = 0;
    __syncthreads();

    // --- greedy NMS over sorted order (== reference's repeated argmax) ---
    float* dst = out + (size_t)n * POST_NMS * 5;
    for (int i = 0; i < PRE_NMS; ++i) {
        if (kept >= POST_NMS) break;                     // uniform (LDS, post-barrier)
        bool alive = (srm[i] == 0) && (ssc[i] > 0.5f * NEG_F);
        if (alive) {
            float bx1 = sx1[i], by1 = sy1[i], bx2 = sx2[i], by2 = sy2[i];
            float ba  = (bx2 - bx1 + 1.f) * (by2 - by1 + 1.f);
            if (tid == 0) {
                float* d = dst + (size_t)kept * 5;
                d[0] = bx1; d[1] = by1; d[2] = bx2; d[3] = by2; d[4] = ssc[i];
                kept++;
            }
            for (int jx = i + 1 + tid; jx < PRE_NMS; jx += 1024) {
                if (!srm[jx]) {
                    float xx1 = fmaxf(bx1, sx1[jx]);
                    float yy1 = fmaxf(by1, sy1[jx]);
                    float xx2 = fminf(bx2, sx2[jx]);
                    float yy2 = fminf(by2, sy2[jx]);
                    float inter = fmaxf(xx2 - xx1 + 1.f, 0.f) * fmaxf(yy2 - yy1 + 1.f, 0.f);
                    float aj = (sx2[jx] - sx1[jx] + 1.f) * (sy2[jx] - sy1[jx] + 1.f);
                    float iou = inter / (ba + aj - inter);
                    if (iou > NMS_THRESH) srm[jx] = 1;
                }
            }
        }
        __syncthreads();
    }
    __syncthreads();
    // zero-fill exhausted slots (reference emits zeros once scores hit NEG)
    for (int q = kept * 5 + tid; q < POST_NMS * 5; q += 1024) dst[q] = 0.f;
}

// ---------------- host launcher ----------------
extern "C" void kernel_launch(void* const* d_in, const int* in_sizes, int n_in,
                              void* d_out, int out_size, void* d_ws, size_t ws_size,
                              hipStream_t stream) {
    const float* obj  = (const float*)d_in[0];   // [N, A, H, W]
    const float* breg = (const float*)d_in[1];   // [N, A*4, H, W]
    const float* anc  = (const float*)d_in[2];   // [N, M, 4]
    float* out = (float*)d_out;                  // [N, POST_NMS, 5]

    // scratch layout (bytes): hist N*16K*4 = 512KB | thr N*16 | cnt N*8 | pairs N*2048*8 = 128KB
    uint32_t* hist  = (uint32_t*)d_ws;
    uint32_t* thr   = hist + (size_t)N_IMG * NBINS;
    uint32_t* cnt   = thr + (size_t)N_IMG * 4;
    uint2*    pairs = (uint2*)(cnt + (size_t)N_IMG * 2);   // byte offset 524480, 8-aligned

    const int zero_n = N_IMG * NBINS + N_IMG * 4 + N_IMG * 2;
    rpn_k0_zero<<<256, 256, 0, stream>>>(hist, zero_n);
    rpn_k1_hist<<<dim3(160, N_IMG), 256, 0, stream>>>(obj, hist);
    rpn_k2_thresh<<<N_IMG, 256, 0, stream>>>(hist, thr);
    rpn_k3_select<<<dim3(160, N_IMG), 256, 0, stream>>>(obj, thr, cnt, pairs);
    rpn_k4_sort_decode_nms<<<N_IMG, 1024, 0, stream>>>(breg, anc, pairs, out);
}